// ArcFaceLoss_51041391346059
// MI455X (gfx1250) — compile-verified
//
#include <hip/hip_runtime.h>
#include <math.h>

typedef float v4f __attribute__((ext_vector_type(4)));
typedef int   v4i __attribute__((ext_vector_type(4)));

#define TPB    256
#define NCHUNK 16   // must be a power of two; gridDim.x == NCHUNK

// ---- ArcFace constants (double-evaluated, stored as f32) ----
#define K_S      64.0f
#define K_COS_M  0.8775825618903728f     // cos(0.5)
#define K_SIN_M  0.479425538604203f      // sin(0.5)
#define K_TH    (-0.8775825618903728f)   // cos(pi-0.5)
#define K_MM     0.23971276930210156f    // sin(pi-0.5)*0.5
#define K_LO    (-1.0f + 1e-7f)
#define K_HI    ( 1.0f - 1e-7f)
#define K_LN2    0.6931471805599453f
#define K_A      92.33248261689366f      // 64*log2(e); exp(64c-64) == exp2(c*K_A - K_A)

__device__ __forceinline__ float fast_exp2(float x) {
#if defined(__HIP_DEVICE_COMPILE__) && __has_builtin(__builtin_amdgcn_exp2f)
  return __builtin_amdgcn_exp2f(x);
#else
  return exp2f(x);
#endif
}
__device__ __forceinline__ float fast_log2(float x) {
#if defined(__HIP_DEVICE_COMPILE__) && __has_builtin(__builtin_amdgcn_logf)
  return __builtin_amdgcn_logf(x);
#else
  return log2f(x);
#endif
}

// exp(S*clip(c) - 64) via hardware exp2, one fma after the clamp
__device__ __forceinline__ float expterm(float c) {
  c = fminf(fmaxf(c, K_LO), K_HI);
  return fast_exp2(fmaf(c, K_A, -K_A));
}

#if defined(__HIP_DEVICE_COMPILE__) &&                                   \
    __has_builtin(__builtin_amdgcn_global_load_async_to_lds_b128) &&     \
    __has_builtin(__builtin_amdgcn_s_wait_asynccnt)
#define USE_ASYNC_LDS 1
typedef __attribute__((address_space(1))) v4i as1_v4i;   // global int4*
typedef __attribute__((address_space(3))) v4i as3_v4i;   // LDS int4*
#endif

// Kernel 1: per-(row,chunk) partial sums of exp(S*c - 64), streamed B128.
__global__ __launch_bounds__(TPB) void arcface_partial(
    const float* __restrict__ cosine, float* __restrict__ part, int C)
{
  const int row   = blockIdx.y;
  const int chunk = blockIdx.x;
  const int tid   = threadIdx.x;
  const int nvec  = C >> 2;                         // C % 4 == 0
  const int per   = (nvec + NCHUNK - 1) / NCHUNK;   // compile-time divisor -> shift
  const int b0    = chunk * per;
  const int e0    = min(b0 + per, nvec);

  const v4f* __restrict__ rowp = (const v4f*)(cosine + (size_t)row * (size_t)C);

  float acc = 0.0f;

#ifdef USE_ASYNC_LDS
  // Double-buffered async global->LDS pipeline (ASYNCcnt).
  // Load index is clamped (never predicated) so every lane of every wave
  // issues exactly one async op per stage -> wait(1)/wait(0) are exact.
  __shared__ v4f stage[2][TPB];
  const int n_it = (e0 - b0 + TPB - 1) / TPB;
  if (n_it > 0) {
    int i0 = min(b0 + tid, e0 - 1);
    __builtin_amdgcn_global_load_async_to_lds_b128(
        (as1_v4i*)(unsigned long long)(const void*)(rowp + i0),
        (as3_v4i*)&stage[0][tid], 0, 0);
    for (int it = 0; it < n_it - 1; ++it) {
      int nx = min(b0 + (it + 1) * TPB + tid, e0 - 1);
      __builtin_amdgcn_global_load_async_to_lds_b128(
          (as1_v4i*)(unsigned long long)(const void*)(rowp + nx),
          (as3_v4i*)&stage[(it + 1) & 1][tid], 0, 0);
      __builtin_amdgcn_s_wait_asynccnt(1);       // buffer 'it' is complete
      asm volatile("" ::: "memory");
      int idx = b0 + it * TPB + tid;
      if (idx < e0) {
        v4f v = stage[it & 1][tid];
        acc += expterm(v.x); acc += expterm(v.y);
        acc += expterm(v.z); acc += expterm(v.w);
      }
    }
    __builtin_amdgcn_s_wait_asynccnt(0);
    asm volatile("" ::: "memory");
    int idx = b0 + (n_it - 1) * TPB + tid;
    if (idx < e0) {
      v4f v = stage[(n_it - 1) & 1][tid];
      acc += expterm(v.x); acc += expterm(v.y);
      acc += expterm(v.z); acc += expterm(v.w);
    }
  }
#else
  // Fallback: direct non-temporal B128 streaming loads.
  for (int idx = b0 + tid; idx < e0; idx += TPB) {
    v4f v = __builtin_nontemporal_load(rowp + idx);
    acc += expterm(v.x); acc += expterm(v.y);
    acc += expterm(v.z); acc += expterm(v.w);
  }
#endif

  __shared__ float red[TPB];
  red[tid] = acc;
  __syncthreads();
  for (int s = TPB / 2; s > 0; s >>= 1) {
    if (tid < s) red[tid] += red[tid + s];
    __syncthreads();
  }
  if (tid == 0) part[row * NCHUNK + chunk] = red[0];
}

// Kernel 2: per-row margin fixup + logsumexp finish + mean (one block, B threads).
__global__ __launch_bounds__(1024) void arcface_finalize(
    const float* __restrict__ cosine, const int* __restrict__ labels,
    const float* __restrict__ part, float* __restrict__ out, int C, int B)
{
  const int r = threadIdx.x;
  __shared__ float red[1024];
  float loss = 0.0f;
  if (r < B) {
    float p = 0.0f;
#pragma unroll
    for (int k = 0; k < NCHUNK; ++k) p += part[r * NCHUNK + k];  // deterministic order
    const int lbl = labels[r];
    float c = cosine[(size_t)r * (size_t)C + lbl];
    c = fminf(fmaxf(c, K_LO), K_HI);
    float sine = sqrtf(fmaxf(1.0f - c * c, 0.0f));
    float phi  = c * K_COS_M - sine * K_SIN_M;
    phi = (c > K_TH) ? phi : (c - K_MM);
    // swap the label column's contribution: -exp(S*c-64) + exp(S*phi-64)
    float pp = p - fast_exp2(fmaf(c, K_A, -K_A)) + fast_exp2(fmaf(phi, K_A, -K_A));
    // logZ = 64 + ln(pp); target logit = S*phi
    loss = 64.0f + fast_log2(pp) * K_LN2 - K_S * phi;
  }
  red[r] = loss;
  __syncthreads();
  for (int s = blockDim.x / 2; s > 0; s >>= 1) {
    if (r < s) red[r] += red[r + s];
    __syncthreads();
  }
  if (r == 0) out[0] = red[0] / (float)B;
}

extern "C" void kernel_launch(void* const* d_in, const int* in_sizes, int n_in,
                              void* d_out, int out_size, void* d_ws, size_t ws_size,
                              hipStream_t stream)
{
  const float* cosine = (const float*)d_in[0];
  const int*   labels = (const int*)d_in[1];
  const int B = in_sizes[1];
  const int C = in_sizes[0] / B;
  float* part = (float*)d_ws;   // B * NCHUNK floats; every slot rewritten each call

  dim3 grid(NCHUNK, B, 1);
  arcface_partial<<<grid, TPB, 0, stream>>>(cosine, part, C);
  arcface_finalize<<<1, B, 0, stream>>>(cosine, labels, part, (float*)d_out, C, B);
}